// graph2linegraph_32049045963020
// MI455X (gfx1250) — compile-verified
//
#include <hip/hip_runtime.h>
#include <stdint.h>

typedef __attribute__((ext_vector_type(4))) float v4f;
typedef __attribute__((ext_vector_type(4))) int   v4i;

#define N_NODES  8192
#define F_DIM    128
#define E_EDGES  16384
#define LG_EDGES 32768

#define NEWX_FLOATS (E_EDGES * 2 * F_DIM)       /* 4,194,304 */
#define LGEI_INTS   (2 * LG_EDGES)              /* 65,536    */
#define NEA_OFFSET  (NEWX_FLOATS + LGEI_INTS)   /* flat float offset of new_edge_attr */

#if __has_builtin(__builtin_amdgcn_global_load_async_to_lds_b128) && \
    __has_builtin(__builtin_amdgcn_s_wait_asynccnt)
#define HAVE_GFX1250_ASYNC 1
#else
#define HAVE_GFX1250_ASYNC 0
#endif

#if HAVE_GFX1250_ASYNC
typedef __attribute__((address_space(1))) v4i glob_v4i;
typedef __attribute__((address_space(3))) v4i lds_v4i;

__device__ __forceinline__ void async_row_b128(const float* g, v4f* l) {
  // AS0 generic -> AS1: identical bit pattern. AS0 generic LDS -> AS3: low 32 bits
  // of a flat LDS address are the LDS offset (aperture lives in the high bits).
  __builtin_amdgcn_global_load_async_to_lds_b128(
      (glob_v4i*)(uint64_t)(uintptr_t)g,
      (lds_v4i*)(uint32_t)(uintptr_t)l,
      0, 0);
}
#endif

// Wave-uniform value assertion: lets the compiler build row base addresses on the
// SALU (s_load/saddr forms) instead of 64-bit VALU add chains per lane.
__device__ __forceinline__ int uniform_i32(int v) {
  return __builtin_amdgcn_readfirstlane(v);
}

// ---- new_x: one wave per edge row i; out[i, 0:128]=.5*(x[src]+ea), out[i,128:256]=.5*(x[tgt]+ea)
__global__ __launch_bounds__(256) void lg_newx_kernel(
    const float* __restrict__ x, const int* __restrict__ edge_index,
    const float* __restrict__ edge_attr, float* __restrict__ out)
{
  const uint32_t gwave = (blockIdx.x * 256u + threadIdx.x) >> 5;  // edge row i
  const uint32_t lane  = threadIdx.x & 31u;
  if (gwave >= E_EDGES) return;
  const uint32_t s = (uint32_t)uniform_i32(edge_index[gwave]);
  const uint32_t t = (uint32_t)uniform_i32(edge_index[E_EDGES + gwave]);

  const v4f* xv4 = (const v4f*)x;          // 32 v4f per row
  const v4f* ev4 = (const v4f*)edge_attr;
  const v4f xs = xv4[s * 32u + lane];
  const v4f xt = xv4[t * 32u + lane];
  const v4f ev = ev4[gwave * 32u + lane];

  v4f o0 = (xs + ev) * 0.5f;
  v4f o1 = (xt + ev) * 0.5f;
  v4f* op = (v4f*)out + gwave * 64u;       // 64 v4f per output row
  __builtin_nontemporal_store(o0, op + lane);
  __builtin_nontemporal_store(o1, op + 32u + lane);
}

// ---- lg_edge_index: i_idx[2i+k]=i ; j_idx[2i+k]=tgt[i]+k*N (row-major nonzero order)
__global__ __launch_bounds__(256) void lg_edges_kernel(
    const int* __restrict__ edge_index, int* __restrict__ out)
{
  const uint32_t tid = blockIdx.x * 256u + threadIdx.x;
  if (tid >= (uint32_t)(2 * LG_EDGES)) return;
  if (tid < (uint32_t)LG_EDGES) {
    out[tid] = (int)(tid >> 1);                         // i_idx
  } else {
    const uint32_t k = tid - LG_EDGES;
    const uint32_t i = k >> 1;
    out[tid] = edge_index[E_EDGES + i] + (int)(k & 1u) * N_NODES;  // j_idx
  }
}

// ---- new_edge_attr: one wave per lg row e; rows 2i and 2i+1 share x[tgt[i]] and ea[i],
// staged once per block into LDS via the gfx1250 async global->LDS path.
__global__ __launch_bounds__(256) void lg_newea_kernel(
    const float* __restrict__ x, const int* __restrict__ edge_index,
    const float* __restrict__ edge_attr, float* __restrict__ out)
{
  const uint32_t w    = threadIdx.x >> 5;        // wave in block: 0..7
  const uint32_t lane = threadIdx.x & 31u;
  const uint32_t e    = blockIdx.x * 8u + w;     // lg row, < 32768
  const uint32_t i    = e >> 1;                  // predecessor edge
  const uint32_t k    = e & 1u;
  const uint32_t t = (uint32_t)uniform_i32(edge_index[E_EDGES + i]); // shared node
  const uint32_t j = t + k * N_NODES;            // successor edge index

  v4f xv, eiv;
#if HAVE_GFX1250_ASYNC
  __shared__ v4f xrow[4][32];
  __shared__ v4f erow[4][32];
  const uint32_t ii = w >> 1;                    // LDS slot, 0..3
  if (k == 0u) {
    async_row_b128(x + (size_t)t * F_DIM + lane * 4u, &xrow[ii][lane]);
    async_row_b128(edge_attr + (size_t)i * F_DIM + lane * 4u, &erow[ii][lane]);
    __builtin_amdgcn_s_wait_asynccnt(0);
  }
  __syncthreads();
  xv  = xrow[ii][lane];
  eiv = erow[ii][lane];
#else
  xv  = ((const v4f*)x)[t * 32u + lane];
  eiv = ((const v4f*)edge_attr)[i * 32u + lane];
#endif
  const v4f ejv = ((const v4f*)edge_attr)[j * 32u + lane];

  v4f o0 = (xv + eiv) * 0.5f;
  v4f o1 = (xv + ejv) * 0.5f;
  v4f* op = (v4f*)out + e * 64u;
  __builtin_nontemporal_store(o0, op + lane);
  __builtin_nontemporal_store(o1, op + 32u + lane);
}

extern "C" void kernel_launch(void* const* d_in, const int* in_sizes, int n_in,
                              void* d_out, int out_size, void* d_ws, size_t ws_size,
                              hipStream_t stream) {
  (void)in_sizes; (void)n_in; (void)out_size; (void)d_ws; (void)ws_size;
  const float* x  = (const float*)d_in[0];
  const int*   ei = (const int*)d_in[1];
  const float* ea = (const float*)d_in[2];

  float* out   = (float*)d_out;
  float* new_x = out;                               // [16384, 256] f32
  int*   lgei  = (int*)(out + NEWX_FLOATS);         // [2, 32768] int32 (raw bits)
  float* nea   = out + NEA_OFFSET;                  // [32768, 256] f32

  lg_newx_kernel <<<E_EDGES / 8, 256, 0, stream>>>(x, ei, ea, new_x);
  lg_edges_kernel<<<(2 * LG_EDGES) / 256, 256, 0, stream>>>(ei, lgei);
  lg_newea_kernel<<<LG_EDGES / 8, 256, 0, stream>>>(x, ei, ea, nea);
}